// Pol2VecMulti_35536559407692
// MI455X (gfx1250) — compile-verified
//
#include <hip/hip_runtime.h>
#include <math.h>

typedef __attribute__((ext_vector_type(2))) float v2f;
typedef __attribute__((ext_vector_type(8))) float v8f;

#define DIMN 16

// ---------------- counting sort of edges by column ----------------

__global__ void k_zero(int* counts, int* cursor, float* out, int ncol) {
    int i = blockIdx.x * blockDim.x + threadIdx.x;
    if (i < ncol) { counts[i] = 0; cursor[i] = 0; }
    if (i == 0) out[0] = 0.0f;
}

__global__ void k_hist(const int* __restrict__ mat_cols, int* counts, int E) {
    int i = blockIdx.x * blockDim.x + threadIdx.x;
    if (i < E) atomicAdd(&counts[mat_cols[i]], 1);
}

// single block of 32 threads: exclusive prefix sum over ncol counts
__global__ void k_scan(const int* __restrict__ counts, int* offsets, int ncol) {
    __shared__ int totals[32];
    int t = threadIdx.x;
    int chunk = (ncol + 31) / 32;
    int begin = t * chunk;
    int endi  = begin + chunk; if (endi > ncol) endi = ncol;
    int run = 0;
    for (int i = begin; i < endi; ++i) { offsets[i] = run; run += counts[i]; }
    totals[t] = run;
    __syncthreads();
    if (t == 0) {
        int acc = 0;
        for (int i = 0; i < 32; ++i) { int v = totals[i]; totals[i] = acc; acc += v; }
        offsets[ncol] = acc;           // == E
    }
    __syncthreads();
    int base = totals[t];
    for (int i = begin; i < endi; ++i) offsets[i] += base;
}

__global__ void k_scatter(const int* __restrict__ mat_cols, const int* __restrict__ offsets,
                          int* cursor, int* perm, int E) {
    int i = blockIdx.x * blockDim.x + threadIdx.x;
    if (i < E) {
        int c = mat_cols[i];
        int p = atomicAdd(&cursor[c], 1);
        perm[offsets[c] + p] = i;
    }
}

// ---------------- main: one wave32 per column, WMMA f32 16x16x4 ----------------

__global__ __launch_bounds__(32)
void k_main(const float* __restrict__ z_rows, const float* __restrict__ z_cols,
            const float* __restrict__ gamma_rows, const float* __restrict__ gamma_cols,
            const float* __restrict__ L, const float* __restrict__ b,
            const float* __restrict__ sigma, const float* __restrict__ col_times,
            const int* __restrict__ mat_rows, const int* __restrict__ yv,
            const int* __restrict__ col_idx_list,
            const int* __restrict__ offsets, const int* __restrict__ perm,
            float* __restrict__ out, int ROWn)
{
    const int c    = blockIdx.x;
    const int lane = threadIdx.x;
    const int half = lane >> 4;      // 0: lanes 0-15, 1: lanes 16-31
    const int lm   = lane & 15;

    const int start = offsets[c];
    const int end   = offsets[c + 1];
    if (start == end) return;        // uniform per wave

    // per-column constants
    const float t   = col_times[c];
    const float tp0 = 1.0f;
    const float tp1 = t;
    const float tp2 = t * t * 0.5f;
    const float tp3 = t * t * t * (1.0f / 6.0f);
    const float gc  = gamma_cols[c];
    const int cidx  = col_idx_list[c];
    const float* Lg = L + (size_t)cidx * (DIMN * DIMN);

    // theta (6 entries), sigma
    float th[6];
    th[0] = -100000.0f;
    th[1] = b[0]; th[2] = b[1]; th[3] = b[2]; th[4] = b[3];
    th[5] = 100000.0f;
    const float inv_s = 1.0f / sigma[0];

    // B fragments for WMMA: Bq[k=2*half+i][n=lm] = Lg[4q+2*half+i][lm]
    v2f Bf[4];
    float2 zc2[4];
#pragma unroll
    for (int q = 0; q < 4; ++q) {
        const int d0 = 4 * q + 2 * half;
        Bf[q].x = Lg[(d0 + 0) * DIMN + lm];
        Bf[q].y = Lg[(d0 + 1) * DIMN + lm];
        zc2[q]  = *(const float2*)(z_cols + (size_t)c * DIMN + d0);
    }

    const size_t vstride = (size_t)ROWn * DIMN;   // stride between order-v slabs of z_rows
    float negll = 0.0f;

    for (int base = start; base < end; base += 16) {
        // ---- build A fragments: A[m=lm][k=4q+2*half+{0,1}] = delta[edge base+lm][d]
        const int  er     = base + lm;
        const bool avalid = (er < end);
        const int  e      = avalid ? perm[er] : 0;
        const int  r      = avalid ? mat_rows[e] : 0;
        const float* zr0  = z_rows + (size_t)r * DIMN;

        v2f Af[4];
#pragma unroll
        for (int q = 0; q < 4; ++q) {
            const int d0 = 4 * q + 2 * half;
            float ax = 0.0f, ay = 0.0f;
            if (avalid) {
                const float* p = zr0 + d0;
                float2 z0 = *(const float2*)(p);
                float2 z1 = *(const float2*)(p + vstride);
                float2 z2 = *(const float2*)(p + 2 * vstride);
                float2 z3 = *(const float2*)(p + 3 * vstride);
                ax = z0.x * tp0 + z1.x * tp1 + z2.x * tp2 + z3.x * tp3 - zc2[q].x;
                ay = z0.y * tp0 + z1.y * tp1 + z2.y * tp2 + z3.y * tp3 - zc2[q].y;
            }
            Af[q].x = ax; Af[q].y = ay;
        }

        // ---- T(16 edges x 16 k) = delta(16x16) * Lg(16x16) via 4 chained K=4 WMMAs
        v8f D = {0.0f, 0.0f, 0.0f, 0.0f, 0.0f, 0.0f, 0.0f, 0.0f};
        D = __builtin_amdgcn_wmma_f32_16x16x4_f32(false, Af[0], false, Bf[0], (short)0, D, false, false);
        D = __builtin_amdgcn_wmma_f32_16x16x4_f32(false, Af[1], false, Bf[1], (short)0, D, false, false);
        D = __builtin_amdgcn_wmma_f32_16x16x4_f32(false, Af[2], false, Bf[2], (short)0, D, false, false);
        D = __builtin_amdgcn_wmma_f32_16x16x4_f32(false, Af[3], false, Bf[3], (short)0, D, false, false);

        // ---- per-edge 2-norm, batched across all 8 edges per half so each
        // reduction stage issues 8 independent ds_bpermute ops (one dscnt wait
        // per stage instead of per shuffle)
        float s2[8];
#pragma unroll
        for (int j = 0; j < 8; ++j) s2[j] = D[j] * D[j];
#pragma unroll
        for (int msk = 1; msk <= 8; msk <<= 1) {
            float tmp[8];
#pragma unroll
            for (int j = 0; j < 8; ++j) tmp[j] = __shfl_xor(s2[j], msk, 16);
#pragma unroll
            for (int j = 0; j < 8; ++j) s2[j] += tmp[j];
        }

        // lane lm (< 8) in each half owns edge m = lm + 8*half; select s2[lm]
        // with cndmasks (no dynamic register indexing -> no scratch)
        float d2 = s2[0];
#pragma unroll
        for (int j = 1; j < 8; ++j) d2 = (lm == j) ? s2[j] : d2;

        if (lm < 8) {
            const int ei = base + lm + 8 * half;
            if (ei < end) {
                const int   ee = perm[ei];
                const int   rr = mat_rows[ee];
                const float f  = -sqrtf(d2) + gamma_rows[rr] + gc;
                const int   yy = yv[ee];     // in [1, 5]
                const float hi = normcdff((th[yy]     - f) * inv_s);
                const float lo = normcdff((th[yy - 1] - f) * inv_s);
                negll -= logf(hi - lo);
            }
        }
    }

    // wave-level reduction of the partial -sum(ll)
    for (int off = 16; off > 0; off >>= 1)
        negll += __shfl_down(negll, off, 32);
    if (lane == 0) atomicAdd(out, negll);
}

// ---------------- launch ----------------

extern "C" void kernel_launch(void* const* d_in, const int* in_sizes, int n_in,
                              void* d_out, int out_size, void* d_ws, size_t ws_size,
                              hipStream_t stream) {
    (void)n_in; (void)out_size; (void)ws_size;
    const float* z_rows       = (const float*)d_in[0];
    const float* z_cols       = (const float*)d_in[1];
    const float* gamma_rows   = (const float*)d_in[2];
    const float* gamma_cols   = (const float*)d_in[3];
    const float* L            = (const float*)d_in[4];
    const float* b            = (const float*)d_in[5];
    const float* sigma        = (const float*)d_in[6];
    const float* col_times    = (const float*)d_in[7];
    const int*   mat_rows     = (const int*)d_in[8];
    const int*   mat_cols     = (const int*)d_in[9];
    const int*   y            = (const int*)d_in[10];
    const int*   col_idx_list = (const int*)d_in[11];

    const int ROWn = in_sizes[2];   // gamma_rows
    const int COLn = in_sizes[7];   // col_times
    const int E    = in_sizes[8];   // mat_rows

    int* w       = (int*)d_ws;
    int* counts  = w;                    // [COLn]
    int* offsets = w + COLn;             // [COLn + 1]
    int* cursor  = w + 2 * COLn + 1;     // [COLn]
    int* perm    = w + 3 * COLn + 1;     // [E]
    float* out   = (float*)d_out;

    k_zero   <<<(COLn + 255) / 256, 256, 0, stream>>>(counts, cursor, out, COLn);
    k_hist   <<<(E    + 255) / 256, 256, 0, stream>>>(mat_cols, counts, E);
    k_scan   <<<1, 32, 0, stream>>>(counts, offsets, COLn);
    k_scatter<<<(E    + 255) / 256, 256, 0, stream>>>(mat_cols, offsets, cursor, perm, E);
    k_main   <<<COLn, 32, 0, stream>>>(z_rows, z_cols, gamma_rows, gamma_cols, L, b, sigma,
                                       col_times, mat_rows, y, col_idx_list,
                                       offsets, perm, out, ROWn);
}